// EmpiricalBayesDistribution_49735721287941
// MI455X (gfx1250) — compile-verified
//
#include <hip/hip_runtime.h>
#include <stdint.h>

#define TILE     128
#define THREADS  128
#define SSTRIDE  7                    // padded float4s per staged sample (bank-conflict: 4-way -> 2-way)
#define REGION   (TILE * SSTRIDE)     // float4s per staged region

// -------------------------------------------------------------------------
// Histogram kernel: stage 4 tensor tiles into LDS with CDNA5 async loads
// (padded LDS layout via the decoupled LDS-address VGPR), then each thread
// packs one sample's 12 base-4 digits (2 bits each) into a 24-bit bin id for
// each of the K=4 points and bumps the bin with a u32 global atomic.
// -------------------------------------------------------------------------
__global__ __launch_bounds__(THREADS) void hist_kernel(
    const float4* __restrict__ in4,    // (C,6,4) as C*6 float4
    const float4* __restrict__ bin4,   // bias_input, same shape
    const float4* __restrict__ out4,   // (C,6,4)
    const float4* __restrict__ bout4,  // bias_output
    unsigned int* __restrict__ hist,   // 4^12 bins (u32 counts)
    int C)
{
    __shared__ float4 smem[4 * REGION];   // 56 KB: in | bias_in | out | bias_out
    const int tid = threadIdx.x;
    const int c0  = blockIdx.x * TILE;

    // number of valid (unpadded) float4s in this tile (handles tail block)
    int limit = (C - c0) * 6;
    if (limit > TILE * 6) limit = TILE * 6;

    const float4* gb[4] = { in4, bin4, out4, bout4 };

    if (limit == TILE * 6) {
        // ---- full tile: straight-line, unguarded async staging ----
#pragma unroll
        for (int r = 0; r < 4; ++r) {
            const float4* base = gb[r] + (size_t)c0 * 6;              // uniform -> SGPR pair
            uint32_t lds_r = (uint32_t)(uintptr_t)(&smem[r * REGION]);
#pragma unroll
            for (int i = 0; i < 6; ++i) {
                uint32_t e = (uint32_t)(i * THREADS + tid);           // float4 index (0..767)
                uint32_t voff  = e * 16u;                             // global byte offset (VGPR)
                uint32_t laddr = lds_r + (e + e / 6u) * 16u;          // padded LDS byte address
                asm volatile("global_load_async_to_lds_b128 %0, %1, %2"
                             :
                             : "v"(laddr), "v"(voff), "s"(base)
                             : "memory");
            }
        }
    } else {
        // ---- tail tile: per-lane guarded staging ----
#pragma unroll
        for (int r = 0; r < 4; ++r) {
            const float4* base = gb[r] + (size_t)c0 * 6;
            uint32_t lds_r = (uint32_t)(uintptr_t)(&smem[r * REGION]);
#pragma unroll
            for (int i = 0; i < 6; ++i) {
                int e = i * THREADS + tid;
                if (e < limit) {
                    uint32_t ue    = (uint32_t)e;
                    uint32_t voff  = ue * 16u;
                    uint32_t laddr = lds_r + (ue + ue / 6u) * 16u;
                    asm volatile("global_load_async_to_lds_b128 %0, %1, %2"
                                 :
                                 : "v"(laddr), "v"(voff), "s"(base)
                                 : "memory");
                }
            }
        }
    }
    asm volatile("s_wait_asynccnt 0x0" ::: "memory");
    __syncthreads();

    const float4* sIn   = &smem[0];
    const float4* sBin  = &smem[REGION];
    const float4* sOut  = &smem[2 * REGION];
    const float4* sBout = &smem[3 * REGION];

    const int c = c0 + tid;
    if (c < C) {
        uint32_t l0 = 0, l1 = 0, l2 = 0, l3 = 0;
#pragma unroll
        for (int h = 0; h < 6; ++h) {                 // input digits, MSB first
            float4 a = sIn[tid * SSTRIDE + h];
            float4 b = sBin[tid * SSTRIDE + h];
            int dx = (int)(a.x + b.x); dx = dx < 0 ? 0 : (dx > 3 ? 3 : dx);
            int dy = (int)(a.y + b.y); dy = dy < 0 ? 0 : (dy > 3 ? 3 : dy);
            int dz = (int)(a.z + b.z); dz = dz < 0 ? 0 : (dz > 3 ? 3 : dz);
            int dw = (int)(a.w + b.w); dw = dw < 0 ? 0 : (dw > 3 ? 3 : dw);
            l0 = (l0 << 2) | (uint32_t)dx;
            l1 = (l1 << 2) | (uint32_t)dy;
            l2 = (l2 << 2) | (uint32_t)dz;
            l3 = (l3 << 2) | (uint32_t)dw;
        }
#pragma unroll
        for (int f = 0; f < 6; ++f) {                 // output digits, LSB last
            float4 a = sOut[tid * SSTRIDE + f];
            float4 b = sBout[tid * SSTRIDE + f];
            int dx = (int)(a.x + b.x); dx = dx < 0 ? 0 : (dx > 3 ? 3 : dx);
            int dy = (int)(a.y + b.y); dy = dy < 0 ? 0 : (dy > 3 ? 3 : dy);
            int dz = (int)(a.z + b.z); dz = dz < 0 ? 0 : (dz > 3 ? 3 : dz);
            int dw = (int)(a.w + b.w); dw = dw < 0 ? 0 : (dw > 3 ? 3 : dw);
            l0 = (l0 << 2) | (uint32_t)dx;
            l1 = (l1 << 2) | (uint32_t)dy;
            l2 = (l2 << 2) | (uint32_t)dz;
            l3 = (l3 << 2) | (uint32_t)dw;
        }
        // 64 MB histogram is L2-resident (192 MB L2); ~0.12 hits/bin contention.
        atomicAdd(&hist[l0], 1u);
        atomicAdd(&hist[l1], 1u);
        atomicAdd(&hist[l2], 1u);
        atomicAdd(&hist[l3], 1u);
    }
}

// -------------------------------------------------------------------------
// In-place convert u32 counts -> float count / C  (matches reference exactly:
// counts are exact integers < 2^24, then one float division).
// -------------------------------------------------------------------------
__global__ __launch_bounds__(256) void finalize_kernel(
    unsigned int* __restrict__ hist, float Cf, int n4)
{
    int i = blockIdx.x * blockDim.x + threadIdx.x;
    if (i < n4) {
        uint4 cv = ((const uint4*)hist)[i];
        float4 o;
        o.x = (float)cv.x / Cf;
        o.y = (float)cv.y / Cf;
        o.z = (float)cv.z / Cf;
        o.w = (float)cv.w / Cf;
        ((float4*)hist)[i] = o;
    }
}

extern "C" void kernel_launch(void* const* d_in, const int* in_sizes, int n_in,
                              void* d_out, int out_size, void* d_ws, size_t ws_size,
                              hipStream_t stream) {
    // setup_inputs order: input_tensor, output_tensor, bias_input, bias_output
    const float4* in4   = (const float4*)d_in[0];
    const float4* out4  = (const float4*)d_in[1];
    const float4* bin4  = (const float4*)d_in[2];
    const float4* bout4 = (const float4*)d_in[3];

    const int C = in_sizes[0] / 24;   // (C,6,4) floats

    // zero the 4^12-bin histogram every call (graph-capturable memset node)
    hipMemsetAsync(d_out, 0, (size_t)out_size * sizeof(float), stream);

    const int nblocks = (C + TILE - 1) / TILE;
    hist_kernel<<<nblocks, THREADS, 0, stream>>>(
        in4, bin4, out4, bout4, (unsigned int*)d_out, C);

    const int n4 = out_size / 4;
    finalize_kernel<<<(n4 + 255) / 256, 256, 0, stream>>>(
        (unsigned int*)d_out, (float)C, n4);
}